// BertCrfForTokenClassification_49392123904212
// MI455X (gfx1250) — compile-verified
//
#include <hip/hip_runtime.h>
#include <hip/hip_bf16.h>

// Problem constants (BertCrfForTokenClassification reference)
#define BB 64
#define TT 512
#define HH 768
#define NK 9
#define NPAD 16           // K=9 padded to one 16-wide WMMA tile
#define MTOT (BB * TT)    // 32768 rows
#define TILES (MTOT / 16) // 2048 M-tiles

typedef float v2f __attribute__((ext_vector_type(2)));
typedef float v8f __attribute__((ext_vector_type(8)));

// ---------------------------------------------------------------------------
// Kernel 1: emissions = hidden @ W + b   via V_WMMA_F32_16X16X4_F32
//   grid 256 blocks x 256 threads (8 waves); one 16x16 tile per wave.
//   W staged transposed + zero-padded in LDS: WldsT[n][kk], n in [0,16).
// ---------------------------------------------------------------------------
__global__ __launch_bounds__(256)
void emissions_wmma_kernel(const float* __restrict__ hidden,
                           const float* __restrict__ W,
                           const float* __restrict__ bias,
                           float* __restrict__ emis /* [MTOT][NPAD] */) {
    __shared__ float WldsT[NPAD * HH]; // 48 KB: WldsT[n*HH + kk] = W[kk*NK + n]

    for (int i = threadIdx.x; i < NPAD * HH; i += blockDim.x) {
        const int n  = i / HH;
        const int kk = i - n * HH;
        WldsT[i] = (n < NK) ? W[kk * NK + n] : 0.0f;
    }
    __syncthreads();

    const int lane = threadIdx.x & 31;
    const int wid  = threadIdx.x >> 5;
    const int tile = blockIdx.x * 8 + wid;        // 256*8 == TILES
    const int row0 = tile * 16;

    const int lo   = lane & 15;        // M index (A) / N index (B,C)
    const int hi2  = (lane >> 4) * 2;  // K sub-offset {0,2}

    const float* aptr = hidden + (size_t)(row0 + lo) * HH + hi2;
    const float* bptr = WldsT + lo * HH + hi2;

    // CDNA5 prefetch path: pull the start of this wave's row block toward L2.
    __builtin_prefetch(aptr, 0, 0);

    v8f c = {};
    #pragma unroll 8
    for (int k = 0; k < HH; k += 4) {
        // A 16x4 f32: v0 = K=(k+hi2), v1 = K=(k+hi2+1), per ISA layout.
        v2f a = *(const v2f*)(aptr + k);
        // B 4x16 f32 from LDS (transposed so the pair is contiguous).
        v2f b = *(const v2f*)(bptr + k);
        c = __builtin_amdgcn_wmma_f32_16x16x4_f32(
                /*neg_a=*/false, a, /*neg_b=*/false, b,
                /*c_mod=*/(short)0, c, /*reuse_a=*/false, /*reuse_b=*/false);
    }

    const float bn = (lo < NK) ? bias[lo] : 0.0f;
    const int mbase = row0 + ((lane >> 4) * 8);
    #pragma unroll
    for (int v = 0; v < 8; ++v) {
        emis[(size_t)(mbase + v) * NPAD + lo] = c[v] + bn;
    }
}

// ---------------------------------------------------------------------------
// Kernel 2: per-batch CRF numerator + forward (logsumexp) scan.
//   grid BB blocks x 32 threads (one wave32 per batch row).
//   Lane j (< 9) owns alpha[j]; trans column j lives in 9 registers;
//   alpha broadcast via wave shuffles — no LDS, no barriers.
// ---------------------------------------------------------------------------
__global__ __launch_bounds__(32)
void crf_scan_kernel(const float* __restrict__ emis /* [MTOT][NPAD] */,
                     const int*   __restrict__ attn,
                     const int*   __restrict__ labels,
                     const float* __restrict__ trans,
                     const float* __restrict__ startT,
                     const float* __restrict__ endT,
                     float* __restrict__ llh /* [BB] */) {
    const int b    = blockIdx.x;
    const int lane = threadIdx.x;
    const size_t eb = (size_t)b * TT * NPAD;

    // ---- Phase A: numerator (strided over t, shuffle-reduced) ----
    float num = 0.0f;
    int   cnt = 0;
    for (int t = lane; t < TT; t += 32) {
        const int lab = labels[b * TT + t];
        const int tg  = (lab == -100) ? 0 : lab;
        const int mk  = attn[b * TT + t];
        cnt += mk;
        if (t == 0) {
            num += startT[tg] + emis[eb + tg];
        } else if (mk) {
            const int labp = labels[b * TT + t - 1];
            const int tp   = (labp == -100) ? 0 : labp;
            num += trans[tp * NK + tg] + emis[eb + (size_t)t * NPAD + tg];
        }
    }
    #pragma unroll
    for (int off = 16; off > 0; off >>= 1) {
        num += __shfl_xor(num, off);
        cnt += __shfl_xor(cnt, off);
    }
    const int last_idx = cnt - 1;
    const int lastlab  = labels[b * TT + last_idx];
    num += endT[(lastlab == -100) ? 0 : lastlab];

    // ---- Phase B: forward algorithm ----
    const bool act = (lane < NK);
    const float NEG = -__builtin_inff();
    float alpha = act ? (startT[lane] + emis[eb + lane]) : NEG;

    float tc[NK];
    #pragma unroll
    for (int i = 0; i < NK; ++i)
        tc[i] = act ? trans[i * NK + lane] : 0.0f;

    for (int t = 1; t < TT; ++t) {
        const float em = act ? emis[eb + (size_t)t * NPAD + lane] : 0.0f;
        const int   mk = attn[b * TT + t];

        float v[NK];
        float mx = NEG;
        #pragma unroll
        for (int i = 0; i < NK; ++i) {
            const float ai = __shfl(alpha, i);
            v[i] = ai + tc[i];
            mx = fmaxf(mx, v[i]);
        }
        float s = 0.0f;
        #pragma unroll
        for (int i = 0; i < NK; ++i) s += expf(v[i] - mx);
        const float nxt = mx + logf(s) + em;
        if (mk) alpha = nxt;  // uniform per block
    }

    // ---- denominator: logsumexp over 9 lanes ----
    float val = act ? (alpha + endT[lane]) : NEG;
    float mx = val;
    #pragma unroll
    for (int off = 16; off > 0; off >>= 1) mx = fmaxf(mx, __shfl_xor(mx, off));
    float s = act ? expf(val - mx) : 0.0f;
    #pragma unroll
    for (int off = 16; off > 0; off >>= 1) s += __shfl_xor(s, off);
    const float denom = mx + logf(s);

    if (lane == 0) llh[b] = num - denom;
}

// ---------------------------------------------------------------------------
// Kernel 3: loss = -mean(llh)   (deterministic, no atomics)
// ---------------------------------------------------------------------------
__global__ __launch_bounds__(32)
void finalize_kernel(const float* __restrict__ llh, float* __restrict__ out) {
    const int lane = threadIdx.x;
    float s = 0.0f;
    for (int i = lane; i < BB; i += 32) s += llh[i];
    #pragma unroll
    for (int off = 16; off > 0; off >>= 1) s += __shfl_xor(s, off);
    if (lane == 0) out[0] = -s / (float)BB;
}

extern "C" void kernel_launch(void* const* d_in, const int* in_sizes, int n_in,
                              void* d_out, int out_size, void* d_ws, size_t ws_size,
                              hipStream_t stream) {
    const float* hidden = (const float*)d_in[0];
    const int*   attn   = (const int*)d_in[1];
    const int*   labels = (const int*)d_in[2];
    const float* W      = (const float*)d_in[3];
    const float* bias   = (const float*)d_in[4];
    const float* trans  = (const float*)d_in[5];
    const float* startT = (const float*)d_in[6];
    const float* endT   = (const float*)d_in[7];
    float* out = (float*)d_out;

    float* ws   = (float*)d_ws;
    float* emis = ws;                              // MTOT*NPAD floats (2 MB)
    float* llh  = ws + (size_t)MTOT * NPAD;        // BB floats

    emissions_wmma_kernel<<<TILES / 8, 256, 0, stream>>>(hidden, W, bias, emis);
    crf_scan_kernel<<<BB, 32, 0, stream>>>(emis, attn, labels, trans, startT, endT, llh);
    finalize_kernel<<<1, 32, 0, stream>>>(llh, out);
}